// RPL_CBAM_44899588113004
// MI455X (gfx1250) — compile-verified
//
#include <hip/hip_runtime.h>
#include <hip/hip_bf16.h>

typedef __attribute__((ext_vector_type(2))) float v2f;
typedef __attribute__((ext_vector_type(8))) float v8f;

#define B_   32
#define C_   256
#define MID_ 32
#define HW_  3136   // 56*56
#define W_   56

__device__ __forceinline__ float sigmoidf_(float v) {
    return 1.0f / (1.0f + __expf(-v));
}

// ---------------------------------------------------------------------------
// Kernel 1: global average pool over H,W per (b,c). Block per (b,c) row.
// ---------------------------------------------------------------------------
__global__ __launch_bounds__(256) void cbam_pool(const float* __restrict__ x,
                                                 float* __restrict__ pooled) {
    const int bc = blockIdx.x;                       // 0..8191
    const float4* row = (const float4*)(x + (size_t)bc * HW_);
    float s = 0.0f;
    for (int i = threadIdx.x; i < HW_ / 4; i += 256) {
        float4 v = row[i];
        s += (v.x + v.y) + (v.z + v.w);
    }
    __shared__ float red[256];
    red[threadIdx.x] = s;
    __syncthreads();
    for (int off = 128; off >= 32; off >>= 1) {
        if (threadIdx.x < off) red[threadIdx.x] += red[threadIdx.x + off];
        __syncthreads();
    }
    if (threadIdx.x < 32) {
        float v = red[threadIdx.x];
        #pragma unroll
        for (int off = 16; off > 0; off >>= 1) v += __shfl_down(v, off, 32);
        if (threadIdx.x == 0) pooled[bc] = v * (1.0f / (float)HW_);
    }
}

// ---------------------------------------------------------------------------
// Kernel 2: per-pixel channel mean & max. Thread per (b, pixel).
// Each c-iteration: a wave reads 32 consecutive floats (coalesced line).
// ---------------------------------------------------------------------------
__global__ __launch_bounds__(256) void cbam_chanstats(const float* __restrict__ x,
                                                      float* __restrict__ avg,
                                                      float* __restrict__ mx) {
    const int gid = blockIdx.x * 256 + threadIdx.x;  // 0..100351
    const int b = gid / HW_;
    const int p = gid - b * HW_;
    const float* base = x + (size_t)b * C_ * HW_ + p;
    float s = 0.0f;
    float m = -3.402823466e38f;
    #pragma unroll 4
    for (int c = 0; c < C_; ++c) {
        float v = base[(size_t)c * HW_];
        s += v;
        m = fmaxf(m, v);
    }
    avg[gid] = s * (1.0f / (float)C_);
    mx[gid]  = m;
}

// ---------------------------------------------------------------------------
// Kernel 3: channel-attention MLP via fp32 WMMA (one wave, EXEC all-1s).
//   h  = relu(pooled[32,256] @ w1^T[256,32])      -> LDS re-layout
//   ca = sigmoid(h[32,32]    @ w2^T[32,256])
// A layout (16x4 f32): lane l: M=l&15, a.{x,y} = A[M][2*(l>>4) + {0,1}]
// B layout (4x16 f32): lane l: N=l&15, b.{x,y} = B[2*(l>>4)+{0,1}][N]
// C/D layout: VGPR i: M = i + 8*(l>>4), N = l&15
// ---------------------------------------------------------------------------
__global__ __launch_bounds__(32) void cbam_mlp_wmma(const float* __restrict__ pooled,
                                                    const float* __restrict__ w1,
                                                    const float* __restrict__ w2,
                                                    float* __restrict__ ca) {
    __shared__ float h_lds[32][32];
    const int lane = threadIdx.x;
    const int half = lane >> 4;      // 0 or 1
    const int l15  = lane & 15;

    // GEMM1: h[b][m] = sum_c pooled[b][c] * w1[m][c]
    for (int bt = 0; bt < 2; ++bt) {
        for (int mt = 0; mt < 2; ++mt) {
            v8f acc = {};
            for (int kk = 0; kk < C_; kk += 4) {
                const int kb = kk + 2 * half;
                v2f a, bm;
                a.x  = pooled[(bt * 16 + l15) * C_ + kb];
                a.y  = pooled[(bt * 16 + l15) * C_ + kb + 1];
                bm.x = w1[(mt * 16 + l15) * C_ + kb];      // B[k][n] = w1[n][k]
                bm.y = w1[(mt * 16 + l15) * C_ + kb + 1];
                acc = __builtin_amdgcn_wmma_f32_16x16x4_f32(
                    false, a, false, bm, (short)0, acc, false, false);
            }
            #pragma unroll
            for (int i = 0; i < 8; ++i) {
                const int brow = bt * 16 + i + 8 * half;
                h_lds[brow][mt * 16 + l15] = fmaxf(acc[i], 0.0f);  // relu
            }
        }
    }
    __syncthreads();

    // GEMM2: ca[b][c] = sigmoid( sum_m h[b][m] * w2[c][m] )
    for (int bt = 0; bt < 2; ++bt) {
        for (int ct = 0; ct < C_ / 16; ++ct) {
            v8f acc = {};
            for (int kk = 0; kk < MID_; kk += 4) {
                const int kb = kk + 2 * half;
                v2f a, bm;
                a.x  = h_lds[bt * 16 + l15][kb];
                a.y  = h_lds[bt * 16 + l15][kb + 1];
                bm.x = w2[(ct * 16 + l15) * MID_ + kb];    // B[k][n] = w2[n][k]
                bm.y = w2[(ct * 16 + l15) * MID_ + kb + 1];
                acc = __builtin_amdgcn_wmma_f32_16x16x4_f32(
                    false, a, false, bm, (short)0, acc, false, false);
            }
            #pragma unroll
            for (int i = 0; i < 8; ++i) {
                const int brow = bt * 16 + i + 8 * half;
                const int ccol = ct * 16 + l15;
                ca[brow * C_ + ccol] = sigmoidf_(acc[i]);
            }
        }
    }
}

// ---------------------------------------------------------------------------
// Kernel 4: spatial attention 7x7 conv (pad 3) on [avg; max] + sigmoid.
// Thread per (b, pixel). Inputs are ~800KB total -> L2-hot.
// ---------------------------------------------------------------------------
__global__ __launch_bounds__(256) void cbam_saconv(const float* __restrict__ avg,
                                                   const float* __restrict__ mx,
                                                   const float* __restrict__ w_sa,
                                                   float* __restrict__ sa) {
    __shared__ float wk[98];
    if (threadIdx.x < 98) wk[threadIdx.x] = w_sa[threadIdx.x];
    __syncthreads();

    const int gid = blockIdx.x * 256 + threadIdx.x;  // 0..100351
    const int b = gid / HW_;
    const int p = gid - b * HW_;
    const int h = p / W_;
    const int w = p - h * W_;
    const float* ba = avg + (size_t)b * HW_;
    const float* bm = mx  + (size_t)b * HW_;

    float acc = 0.0f;
    #pragma unroll
    for (int kh = 0; kh < 7; ++kh) {
        const int hh = h + kh - 3;
        if (hh < 0 || hh >= W_) continue;
        #pragma unroll
        for (int kw = 0; kw < 7; ++kw) {
            const int ww = w + kw - 3;
            if (ww < 0 || ww >= W_) continue;
            const int q = hh * W_ + ww;
            acc = fmaf(ba[q], wk[kh * 7 + kw], acc);
            acc = fmaf(bm[q], wk[49 + kh * 7 + kw], acc);
        }
    }
    sa[gid] = sigmoidf_(acc);
}

// ---------------------------------------------------------------------------
// Kernel 5: out = x * (1 + ca[b,c] * sa[b,h,w])   (ALPHA = 1)
// Block per (b,c) row; float4 in/out. This is the 206 MB HBM-bound pass.
// ---------------------------------------------------------------------------
__global__ __launch_bounds__(256) void cbam_fuse(const float* __restrict__ x,
                                                 const float* __restrict__ ca,
                                                 const float* __restrict__ sa,
                                                 float* __restrict__ out) {
    const int bc = blockIdx.x;             // 0..8191
    const int b  = bc >> 8;                // /C_
    const float caw = ca[bc];
    const float4* xr  = (const float4*)(x  + (size_t)bc * HW_);
    const float4* sar = (const float4*)(sa + (size_t)b  * HW_);
    float4* outr      = (float4*)(out + (size_t)bc * HW_);
    for (int i = threadIdx.x; i < HW_ / 4; i += 256) {
        float4 xv = xr[i];
        float4 sv = sar[i];
        float4 o;
        o.x = xv.x * fmaf(caw, sv.x, 1.0f);
        o.y = xv.y * fmaf(caw, sv.y, 1.0f);
        o.z = xv.z * fmaf(caw, sv.z, 1.0f);
        o.w = xv.w * fmaf(caw, sv.w, 1.0f);
        outr[i] = o;
    }
}

// ---------------------------------------------------------------------------
extern "C" void kernel_launch(void* const* d_in, const int* in_sizes, int n_in,
                              void* d_out, int out_size, void* d_ws, size_t ws_size,
                              hipStream_t stream) {
    const float* x    = (const float*)d_in[0];   // [32,256,56,56]
    const float* w1   = (const float*)d_in[1];   // [32,256]
    const float* w2   = (const float*)d_in[2];   // [256,32]
    const float* w_sa = (const float*)d_in[3];   // [1,2,7,7] = 98
    float* out = (float*)d_out;

    // Workspace layout (floats)
    float* ws     = (float*)d_ws;
    float* pooled = ws;                              // 8192
    float* ca     = pooled + B_ * C_;                // 8192
    float* avg    = ca + B_ * C_;                    // 100352
    float* mx     = avg + B_ * HW_;                  // 100352
    float* sa     = mx + B_ * HW_;                   // 100352
    (void)in_sizes; (void)n_in; (void)out_size; (void)ws_size;

    const int bcBlocks = B_ * C_;                    // 8192
    const int pxBlocks = (B_ * HW_) / 256;           // 392 (exact)

    cbam_pool     <<<bcBlocks, 256, 0, stream>>>(x, pooled);
    cbam_chanstats<<<pxBlocks, 256, 0, stream>>>(x, avg, mx);
    cbam_mlp_wmma <<<1,        32,  0, stream>>>(pooled, w1, w2, ca);
    cbam_saconv   <<<pxBlocks, 256, 0, stream>>>(avg, mx, w_sa, sa);
    cbam_fuse     <<<bcBlocks, 256, 0, stream>>>(x, ca, sa, out);
}